// GCN_Net_32418413150680
// MI455X (gfx1250) — compile-verified
//
#include <hip/hip_runtime.h>

#define NNODES 50000
#define NEDGES 800000
#define NPAIRS 262144
#define FIN    572
#define CH1    256
#define CH2    128
#define NCLS   65
#define ETOT   (NEDGES + NNODES)

typedef __attribute__((ext_vector_type(2))) float v2f;
typedef __attribute__((ext_vector_type(8))) float v8f;

// CDNA5 async Global->LDS copy path (ASYNCcnt), probe-guarded.
// Builtin signature (from hipcc diagnostic): params are pointers to
// int __attribute__((vector_size(16))) with AS1 (global) / AS3 (LDS).
#if defined(__gfx1250__) &&                                          \
    __has_builtin(__builtin_amdgcn_global_load_async_to_lds_b128) && \
    __has_builtin(__builtin_amdgcn_s_wait_asynccnt)
#define USE_ASYNC 1
typedef __attribute__((__vector_size__(16))) int b128i;
#define ASYNC_CP16(gp, lp)                                           \
  __builtin_amdgcn_global_load_async_to_lds_b128(                    \
      (__attribute__((address_space(1))) b128i*)(gp),                \
      (__attribute__((address_space(3))) b128i*)(lp), 0, 0)
#else
#define USE_ASYNC 0
#endif

// ------------------------------------------------------------------
// WMMA fp32 GEMM: C[M,N] = A[M,K] @ B[K,N] (+ optional bias[N]).
// Row-major fp32. If gid0 != nullptr, logical A row r / col k is gathered:
//   k <  gsplit : Abase[gid0[r]*gsplit + k]
//   k >= gsplit : Abase[gid1[r]*gsplit + (k-gsplit)]
// Block tile 64x64, 8 waves (4 along M x 2 along N), K-chunk 32.
// Double-buffered LDS; async Global->LDS (B128) prefetch of the next
// K-chunk overlapped with 16 V_WMMA_F32_16X16X4_F32 per wave per chunk.
// ------------------------------------------------------------------
#define BM 64
#define BN 64
#define BK 32
#define APAD 36   // 144B rows: 16B-aligned, b64-friendly fragments
#define BPAD 68   // 272B rows: 16B-aligned

__global__ __launch_bounds__(256) void wmma_gemm_kernel(
    const float* __restrict__ A, const float* __restrict__ B,
    float* __restrict__ C, int M, int N, int K,
    const float* __restrict__ bias,
    const int* __restrict__ gid0, const int* __restrict__ gid1,
    int gsplit, int lda)
{
  __shared__ float As[2][BM][APAD];
  __shared__ float Bs[2][BK][BPAD];

  const int m0   = blockIdx.y * BM;
  const int n0   = blockIdx.x * BN;
  const int tid  = threadIdx.x;
  const int lane = tid & 31;
  const int wave = tid >> 5;
  const int mw   = wave & 3;            // 4 waves along M
  const int nw   = wave >> 2;           // 2 waves along N
  const int l15  = lane & 15;
  const int koff = (lane >> 4) << 1;    // 0 (lanes 0-15) or 2 (lanes 16-31)
  const int rsel = (lane >> 4) << 3;    // C/D row offset: 0 or +8

  v8f acc0 = {0.f, 0.f, 0.f, 0.f, 0.f, 0.f, 0.f, 0.f};
  v8f acc1 = {0.f, 0.f, 0.f, 0.f, 0.f, 0.f, 0.f, 0.f};

  // Stage one 64x32 A tile + 32x64 B tile into LDS buffer `buf`.
  auto stage = [&](int buf, int k0) {
    const bool fullMK = (m0 + BM <= M) && (k0 + BK <= K);
    const bool fullB  = (k0 + BK <= K) && (n0 + BN <= N) && ((N & 3) == 0);

    // ---- A tile: 512 groups of 4 floats, 2 per thread ----
    if (fullMK) {
#pragma unroll
      for (int i = 0; i < 2; ++i) {
        int g  = tid + i * 256;
        int r  = g >> 3;            // 8 groups per 32-float row
        int c4 = (g & 7) * 4;
        const float* gp;
        if (gid0 == nullptr) {
          gp = A + (size_t)(m0 + r) * lda + (k0 + c4);
        } else {
          int gk   = k0 + c4;       // gsplit is a multiple of 4: no straddle
          int node = (gk < gsplit) ? gid0[m0 + r] : gid1[m0 + r];
          int kk   = (gk < gsplit) ? gk : (gk - gsplit);
          gp = A + (size_t)node * gsplit + kk;
        }
        float* lp = &As[buf][r][c4];
#if USE_ASYNC
        ASYNC_CP16(gp, lp);
#else
        *(float4*)lp = *(const float4*)gp;
#endif
      }
    } else {
#pragma unroll
      for (int i = 0; i < (BM * BK) / 256; ++i) {
        int idx = tid + i * 256;
        int r = idx >> 5, c = idx & 31;
        int gr = m0 + r, gk = k0 + c;
        float v = 0.f;
        if (gr < M && gk < K) {
          if (gid0 == nullptr) {
            v = A[(size_t)gr * lda + gk];
          } else {
            int node = (gk < gsplit) ? gid0[gr] : gid1[gr];
            int kk   = (gk < gsplit) ? gk : (gk - gsplit);
            v = A[(size_t)node * gsplit + kk];
          }
        }
        As[buf][r][c] = v;
      }
    }

    // ---- B tile: 512 groups of 4 floats, 2 per thread ----
    if (fullB) {
#pragma unroll
      for (int i = 0; i < 2; ++i) {
        int g  = tid + i * 256;
        int r  = g >> 4;            // 16 groups per 64-float row
        int c4 = (g & 15) * 4;
        const float* gp = B + (size_t)(k0 + r) * N + (n0 + c4);
        float* lp = &Bs[buf][r][c4];
#if USE_ASYNC
        ASYNC_CP16(gp, lp);
#else
        *(float4*)lp = *(const float4*)gp;
#endif
      }
    } else {
#pragma unroll
      for (int i = 0; i < (BK * BN) / 256; ++i) {
        int idx = tid + i * 256;
        int r = idx >> 6, c = idx & 63;
        int gk = k0 + r, gn = n0 + c;
        Bs[buf][r][c] = (gk < K && gn < N) ? B[(size_t)gk * N + gn] : 0.f;
      }
    }
  };

  const int nChunks = (K + BK - 1) / BK;
  stage(0, 0);
#if USE_ASYNC
  __builtin_amdgcn_s_wait_asynccnt(0);
#endif
  __syncthreads();

  for (int ci = 0; ci < nChunks; ++ci) {
    const int cur = ci & 1;
    if (ci + 1 < nChunks) stage(cur ^ 1, (ci + 1) * BK);   // overlapped prefetch

    // ---- 8 k-steps of 4, two 16x16 WMMA tiles per wave per step ----
#pragma unroll
    for (int kk = 0; kk < BK; kk += 4) {
      v2f a = *(const v2f*)&As[cur][mw * 16 + l15][kk + koff];  // aligned b64
      v2f b0, b1;
      b0.x = Bs[cur][kk + koff][nw * 32 + l15];
      b0.y = Bs[cur][kk + koff + 1][nw * 32 + l15];
      b1.x = Bs[cur][kk + koff][nw * 32 + 16 + l15];
      b1.y = Bs[cur][kk + koff + 1][nw * 32 + 16 + l15];
      acc0 = __builtin_amdgcn_wmma_f32_16x16x4_f32(false, a, false, b0,
                                                   (short)0, acc0, false, false);
      acc1 = __builtin_amdgcn_wmma_f32_16x16x4_f32(false, a, false, b1,
                                                   (short)0, acc1, false, false);
    }
#if USE_ASYNC
    __builtin_amdgcn_s_wait_asynccnt(0);   // next tile landed in LDS
#endif
    __syncthreads();
  }

  // ---- store D: VGPR v -> row (v | v+8), col = lane&15 ----
#pragma unroll
  for (int v = 0; v < 8; ++v) {
    int gr  = m0 + mw * 16 + v + rsel;
    int gc0 = n0 + nw * 32 + l15;
    int gc1 = gc0 + 16;
    if (gr < M) {
      if (gc0 < N) C[(size_t)gr * N + gc0] = acc0[v] + (bias ? bias[gc0] : 0.f);
      if (gc1 < N) C[(size_t)gr * N + gc1] = acc1[v] + (bias ? bias[gc1] : 0.f);
    }
  }
}

// ------------------------------------------------------------------
// Attention scores: s[n,h] = dot(H[n, h*256 : h*256+256], a_src[h]), same d.
// One wave per (node, head).
// ------------------------------------------------------------------
__global__ __launch_bounds__(256) void attn_scores_kernel(
    const float* __restrict__ H, const float* __restrict__ a_src,
    const float* __restrict__ a_dst, float* __restrict__ sb,
    float* __restrict__ db, int n)
{
  int wid  = (blockIdx.x * blockDim.x + threadIdx.x) >> 5;
  int lane = threadIdx.x & 31;
  if (wid >= n * 2) return;
  int node = wid >> 1, head = wid & 1;
  const float* hp = H + (size_t)node * 512 + head * 256;
  const float* as = a_src + head * 256;
  const float* ad = a_dst + head * 256;
  float ss = 0.f, dd = 0.f;
  for (int c = lane; c < 256; c += 32) {
    float v = hp[c];
    ss += v * as[c];
    dd += v * ad[c];
  }
  for (int off = 16; off > 0; off >>= 1) {
    ss += __shfl_down(ss, off);
    dd += __shfl_down(dd, off);
  }
  if (lane == 0) { sb[wid] = ss; db[wid] = dd; }
}

// monotone float <-> ordered-uint mapping for atomicMax on floats
__device__ __forceinline__ unsigned fmap(float f) {
  unsigned u = __float_as_uint(f);
  return (u & 0x80000000u) ? ~u : (u | 0x80000000u);
}
__device__ __forceinline__ float funmap(unsigned u) {
  return __uint_as_float((u & 0x80000000u) ? (u & 0x7FFFFFFFu) : ~u);
}
__device__ __forceinline__ float leaky02(float a) {
  return (a > 0.f) ? a : 0.2f * a;
}

__global__ void gat_init_kernel(float* __restrict__ outacc,
                                unsigned* __restrict__ m,
                                float* __restrict__ z, int n)
{
  int t = blockIdx.x * blockDim.x + threadIdx.x;
  if (t < n * 512) outacc[t] = 0.f;
  if (t < n * 2) { m[t] = fmap(-__builtin_huge_valf()); z[t] = 0.f; }
}

__global__ void edge_max_kernel(const int* __restrict__ src,
                                const int* __restrict__ dst,
                                const float* __restrict__ sb,
                                const float* __restrict__ db,
                                unsigned* __restrict__ m, int nE, int nN)
{
  int t = blockIdx.x * blockDim.x + threadIdx.x;
  if (t >= (nE + nN) * 2) return;
  int e = t >> 1, h = t & 1;
  int s_ = (e < nE) ? src[e] : (e - nE);
  int d_ = (e < nE) ? dst[e] : (e - nE);
  float a = leaky02(sb[s_ * 2 + h] + db[d_ * 2 + h]);
  atomicMax(&m[d_ * 2 + h], fmap(a));
}

__global__ void edge_exp_kernel(const int* __restrict__ src,
                                const int* __restrict__ dst,
                                const float* __restrict__ sb,
                                const float* __restrict__ db,
                                const unsigned* __restrict__ m,
                                float* __restrict__ z,
                                float* __restrict__ ebuf, int nE, int nN)
{
  int t = blockIdx.x * blockDim.x + threadIdx.x;
  if (t >= (nE + nN) * 2) return;
  int e = t >> 1, h = t & 1;
  int s_ = (e < nE) ? src[e] : (e - nE);
  int d_ = (e < nE) ? dst[e] : (e - nE);
  float a  = leaky02(sb[s_ * 2 + h] + db[d_ * 2 + h]);
  float ev = expf(a - funmap(m[d_ * 2 + h]));
  ebuf[t] = ev;
  atomicAdd(&z[d_ * 2 + h], ev);
}

// per-edge message scatter: 64 threads per edge, 4 channels x 2 heads each
__global__ void edge_msg_kernel(const int* __restrict__ src,
                                const int* __restrict__ dst,
                                const float* __restrict__ H,
                                const float* __restrict__ ebuf,
                                const float* __restrict__ z,
                                float* __restrict__ outacc, int nE, int nN)
{
  long long t = (long long)blockIdx.x * blockDim.x + threadIdx.x;
  if (t >= (long long)(nE + nN) * 64) return;
  int e  = (int)(t >> 6);
  int c0 = (int)(t & 63);
  int s_ = (e < nE) ? src[e] : (e - nE);
  int d_ = (e < nE) ? dst[e] : (e - nE);
#pragma unroll
  for (int h = 0; h < 2; ++h) {
    float w = ebuf[e * 2 + h] / (z[d_ * 2 + h] + 1e-16f);
    const float* hp = H + (size_t)s_ * 512 + h * 256;
    float* op = outacc + (size_t)d_ * 512 + h * 256;
#pragma unroll
    for (int i = 0; i < 4; ++i) {
      int c = c0 + i * 64;
      atomicAdd(&op[c], w * hp[c]);
    }
  }
}

// mean over heads + bias
__global__ void gat_finalize_kernel(const float* __restrict__ outacc,
                                    const float* __restrict__ bias,
                                    float* __restrict__ out, int n)
{
  int t = blockIdx.x * blockDim.x + threadIdx.x;
  if (t >= n * 256) return;
  int node = t >> 8, c = t & 255;
  out[t] = 0.5f * (outacc[(size_t)node * 512 + c] +
                   outacc[(size_t)node * 512 + 256 + c]) + bias[c];
}

// ---------------- BatchNorm (training-style batch stats) ----------------
__global__ void stats_init_kernel(float* __restrict__ stats, int C)
{
  int t = blockIdx.x * blockDim.x + threadIdx.x;
  if (t < 2 * C) stats[t] = 0.f;
}

// blockDim.x == C (128 or 256); thread = column, block = row chunk
__global__ void col_stats_kernel(const float* __restrict__ X,
                                 float* __restrict__ stats,
                                 int rows, int C, int rowsPerBlock)
{
  int c  = threadIdx.x;
  int r0 = blockIdx.x * rowsPerBlock;
  int r1 = r0 + rowsPerBlock;
  if (r1 > rows) r1 = rows;
  float s = 0.f, s2 = 0.f;
  for (int r = r0; r < r1; ++r) {
    float v = X[(size_t)r * C + c];
    s  += v;
    s2 += v * v;
  }
  atomicAdd(&stats[c], s);
  atomicAdd(&stats[C + c], s2);
}

__global__ void bn_relu_kernel(const float* __restrict__ X,
                               const float* __restrict__ stats,
                               const float* __restrict__ g,
                               const float* __restrict__ b,
                               float* __restrict__ Y, int rows, int C)
{
  long long t = (long long)blockIdx.x * blockDim.x + threadIdx.x;
  if (t >= (long long)rows * C) return;
  int c = (int)(t % C);
  float inv = 1.0f / (float)rows;
  float mu  = stats[c] * inv;
  float var = stats[C + c] * inv - mu * mu;
  float y = g[c] * (X[t] - mu) * rsqrtf(var + 1e-5f) + b[c];
  Y[t] = (y > 0.f) ? y : 0.f;
}

// ==================================================================
extern "C" void kernel_launch(void* const* d_in, const int* in_sizes, int n_in,
                              void* d_out, int out_size, void* d_ws, size_t ws_size,
                              hipStream_t stream)
{
  (void)in_sizes; (void)n_in; (void)out_size; (void)ws_size;

  const int*   edge_index = (const int*)d_in[0];
  const float* x      = (const float*)d_in[1];
  const int*   edge_id= (const int*)d_in[2];
  const float* W1     = (const float*)d_in[3];
  const float* a_src1 = (const float*)d_in[4];
  const float* a_dst1 = (const float*)d_in[5];
  const float* b1     = (const float*)d_in[6];
  const float* bn1_g  = (const float*)d_in[7];
  const float* bn1_b  = (const float*)d_in[8];
  const float* W2     = (const float*)d_in[9];
  const float* a_src2 = (const float*)d_in[10];
  const float* a_dst2 = (const float*)d_in[11];
  const float* b2     = (const float*)d_in[12];
  const float* bn2_g  = (const float*)d_in[13];
  const float* bn2_b  = (const float*)d_in[14];
  const float* lw1    = (const float*)d_in[15];
  const float* lb1    = (const float*)d_in[16];
  const float* bn3_g  = (const float*)d_in[17];
  const float* bn3_b  = (const float*)d_in[18];
  const float* lw2    = (const float*)d_in[19];
  const float* lb2    = (const float*)d_in[20];
  const float* bn4_g  = (const float*)d_in[21];
  const float* bn4_b  = (const float*)d_in[22];
  const float* fw     = (const float*)d_in[23];
  const float* fb     = (const float*)d_in[24];
  float* out = (float*)d_out;

  const int* srcp = edge_index;            // edge_index[0]
  const int* dstp = edge_index + NEDGES;   // edge_index[1]
  const int* eid0 = edge_id;
  const int* eid1 = edge_id + NPAIRS;

  // ---- workspace carve (floats) ----
  float* ws = (float*)d_ws;
  const size_t SLOT = (size_t)NPAIRS * 128;             // 33,554,432 >= NNODES*512
  float*    slotA = ws;                     ws += SLOT; // GEMM out H / z1
  float*    slotB = ws;                     ws += SLOT; // aggregation acc / z2
  float*    h1b   = ws;                     ws += (size_t)NNODES * 256;
  float*    h2b   = ws;                     ws += (size_t)NNODES * 256;
  float*    sbuf  = ws;                     ws += (size_t)NNODES * 2;
  float*    dbuf  = ws;                     ws += (size_t)NNODES * 2;
  unsigned* mbuf  = (unsigned*)ws;          ws += (size_t)NNODES * 2;
  float*    zbuf  = ws;                     ws += (size_t)NNODES * 2;
  float*    ebuf  = ws;                     ws += (size_t)ETOT * 2;
  float*    stats = ws;                     ws += 512;

  auto run_gat = [&](const float* xin, int F, const float* W, const float* as,
                     const float* ad, const float* bias, float* hout) {
    dim3 g1((512 + BN - 1) / BN, (NNODES + BM - 1) / BM);
    wmma_gemm_kernel<<<g1, 256, 0, stream>>>(xin, W, slotA, NNODES, 512, F,
                                             nullptr, nullptr, nullptr, 0, F);
    int waves = NNODES * 2;
    attn_scores_kernel<<<(waves * 32 + 255) / 256, 256, 0, stream>>>(
        slotA, as, ad, sbuf, dbuf, NNODES);
    gat_init_kernel<<<(NNODES * 512 + 255) / 256, 256, 0, stream>>>(
        slotB, mbuf, zbuf, NNODES);
    int et2 = ETOT * 2;
    edge_max_kernel<<<(et2 + 255) / 256, 256, 0, stream>>>(
        srcp, dstp, sbuf, dbuf, mbuf, NEDGES, NNODES);
    edge_exp_kernel<<<(et2 + 255) / 256, 256, 0, stream>>>(
        srcp, dstp, sbuf, dbuf, mbuf, zbuf, ebuf, NEDGES, NNODES);
    long long mthreads = (long long)ETOT * 64;
    edge_msg_kernel<<<(unsigned)((mthreads + 255) / 256), 256, 0, stream>>>(
        srcp, dstp, slotA, ebuf, zbuf, slotB, NEDGES, NNODES);
    gat_finalize_kernel<<<(NNODES * 256 + 255) / 256, 256, 0, stream>>>(
        slotB, bias, hout, NNODES);
  };

  auto run_bn = [&](float* X, const float* g, const float* b, int rows, int C) {
    stats_init_kernel<<<(2 * C + 255) / 256, 256, 0, stream>>>(stats, C);
    const int RPB = 256;
    col_stats_kernel<<<(rows + RPB - 1) / RPB, C, 0, stream>>>(X, stats, rows, C, RPB);
    bn_relu_kernel<<<(unsigned)(((long long)rows * C + 255) / 256), 256, 0, stream>>>(
        X, stats, g, b, X, rows, C);
  };

  // ---- GAT layer 1 ----
  run_gat(x, FIN, W1, a_src1, a_dst1, b1, h1b);
  run_bn(h1b, bn1_g, bn1_b, NNODES, CH1);

  // ---- GAT layer 2 ----
  run_gat(h1b, CH1, W2, a_src2, a_dst2, b2, h2b);
  run_bn(h2b, bn2_g, bn2_b, NNODES, CH1);

  // ---- pair MLP ----
  // z1 = concat(h2[e0], h2[e1]) @ lw1 + lb1  (gathered A, K=512, split 256)
  {
    dim3 g((CH2 + BN - 1) / BN, (NPAIRS + BM - 1) / BM);
    wmma_gemm_kernel<<<g, 256, 0, stream>>>(h2b, lw1, slotA, NPAIRS, CH2, 2 * CH1,
                                            lb1, eid0, eid1, CH1, CH1);
  }
  run_bn(slotA, bn3_g, bn3_b, NPAIRS, CH2);

  // z2 = z1 @ lw2 + lb2
  {
    dim3 g((CH2 + BN - 1) / BN, (NPAIRS + BM - 1) / BM);
    wmma_gemm_kernel<<<g, 256, 0, stream>>>(slotA, lw2, slotB, NPAIRS, CH2, CH2,
                                            lb2, nullptr, nullptr, 0, CH2);
  }
  run_bn(slotB, bn4_g, bn4_b, NPAIRS, CH2);

  // out = z2 @ fw + fb  (N=65, tile guards handle the ragged edge)
  {
    dim3 g((NCLS + BN - 1) / BN, (NPAIRS + BM - 1) / BM);
    wmma_gemm_kernel<<<g, 256, 0, stream>>>(slotB, fw, out, NPAIRS, NCLS, CH2,
                                            fb, nullptr, nullptr, 0, CH2);
  }
}